// RootModule_2216203125290
// MI455X (gfx1250) — compile-verified
//
#include <hip/hip_runtime.h>

typedef __attribute__((ext_vector_type(2))) float v2f;
typedef __attribute__((ext_vector_type(8))) float v8f;
typedef __attribute__((ext_vector_type(4))) int   v4i;

#define B_TOT  8192
#define WIDTH_ 1024
#define NH_    1024
#define RANK_  16
#define L_     4

// ---- CDNA5 async global->LDS path (guarded so we always compile) ----------
#if defined(__has_builtin)
#if __has_builtin(__builtin_amdgcn_global_load_async_to_lds_b128) && \
    __has_builtin(__builtin_amdgcn_s_wait_asynccnt)
#define USE_ASYNC 1
#endif
#endif
#ifndef USE_ASYNC
#define USE_ASYNC 0
#endif

static __device__ __forceinline__ v2f ld2(const float* p) {
  return *reinterpret_cast<const v2f*>(p);
}

// D = A(16x4 f32) * B(4x16 f32) + C, exact fp32 on the matrix pipe.
static __device__ __forceinline__ v8f wmma_f32(v2f a, v2f b, v8f c) {
  return __builtin_amdgcn_wmma_f32_16x16x4_f32(
      /*neg_a=*/false, a, /*neg_b=*/false, b,
      /*c_mod=*/(short)0, c, /*reuse_a=*/false, /*reuse_b=*/false);
}

#if USE_ASYNC
// 16B per-lane async copy global -> LDS, tracked by ASYNCcnt.
// Builtin signature (from clang diagnostic): (AS1 v4i*, AS3 v4i*, imm, imm).
static __device__ __forceinline__ void cp_async16(const float* g, float* lds) {
  __builtin_amdgcn_global_load_async_to_lds_b128(
      (__attribute__((address_space(1))) v4i*)(void*)g,
      (__attribute__((address_space(3))) v4i*)(void*)lds,
      /*imm offset=*/0, /*cpol=*/0);
}
#endif

// ---------------------------------------------------------------------------
// y[b, r] = alpha(b, layer) * sum_k x[b,k] * Bsl[r,k]
// One wave owns one 16-row M tile; N = RANK = one 16-wide tile; K = WIDTH.
// Alpha affine map (nu, t, alpha_W row, alpha_b) fused into the epilogue.
// ---------------------------------------------------------------------------
__global__ __launch_bounds__(256, 1)
void lowrank_kernel(const float* __restrict__ x,    // B x WIDTH
                    const float* __restrict__ Bsl,  // RANK x WIDTH (layer i)
                    const float* __restrict__ nu,   // B x 3
                    const float* __restrict__ tt,   // B
                    const float* __restrict__ aWl,  // 4 floats (alpha_W[i,:])
                    const float* __restrict__ abl,  // 1 float  (alpha_b[i])
                    float* __restrict__ y)          // B x RANK
{
  const int lane  = threadIdx.x & 31;
  const int wave  = threadIdx.x >> 5;
  const int hh    = lane >> 4;    // half-wave select
  const int l16   = lane & 15;
  const int mbase = blockIdx.x * 128 + wave * 16;

  const float* xrow = x   + (size_t)(mbase + l16) * WIDTH_;  // A frag: M = l16
  const float* brow = Bsl + (size_t)l16 * WIDTH_;            // B frag: N(=r) = l16

  v8f acc = (v8f)0.0f;
#pragma unroll 8
  for (int k = 0; k < WIDTH_; k += 4) {
    const int k0 = k + 2 * hh;          // K pair {k0, k0+1}, contiguous float2
    acc = wmma_f32(ld2(xrow + k0), ld2(brow + k0), acc);
  }

  const float a0 = aWl[0], a1 = aWl[1], a2 = aWl[2], a3 = aWl[3], ab = abl[0];
#pragma unroll
  for (int v = 0; v < 8; ++v) {
    const int row = mbase + v + 8 * hh;          // C layout: M = v + 8*half
    const float alpha = nu[row * 3 + 0] * a0 + nu[row * 3 + 1] * a1 +
                        nu[row * 3 + 2] * a2 + tt[row] * a3 + ab;
    y[(size_t)row * RANK_ + l16] = alpha * acc[v];
  }
}

// ---------------------------------------------------------------------------
// out[b,n] = sum_k x[b,k]*Wl[n,k] + sum_r y[b,r]*Al[n,r] + bias[n]
// 128x128 block tile. K staged via double-buffered LDS slabs (32 wide),
// filled with GLOBAL_LOAD_ASYNC_TO_LDS_B128 (ASYNCcnt) when available.
// 8 waves: wave tile 32x64 -> 2x4 accumulators of 16x16.
// ---------------------------------------------------------------------------
__global__ __launch_bounds__(256, 1)
void layer_gemm_kernel(const float* __restrict__ x,     // B x WIDTH
                       const float* __restrict__ Wl,    // NH x WIDTH (layer i)
                       const float* __restrict__ y,     // B x RANK (alpha-scaled)
                       const float* __restrict__ Al,    // NH x RANK (layer i)
                       const float* __restrict__ bias,  // NH
                       float* __restrict__ out)         // B x NH
{
  __shared__ float sX[2][128][36];   // 36-float stride keeps rows 16B aligned
  __shared__ float sW[2][128][36];   // ~72 KB total: WGP has 320 KB

  const int tid  = threadIdx.x;
  const int lane = tid & 31;
  const int wave = tid >> 5;
  const int hh   = lane >> 4;
  const int l16  = lane & 15;
  const int mw   = (wave & 3) * 32;   // wave M offset inside block tile
  const int nw   = (wave >> 2) * 64;  // wave N offset inside block tile
  const int bm   = blockIdx.x * 128;
  const int bn   = blockIdx.y * 128;
  const int lrow = tid >> 3;          // 0..31   (tile staging)
  const int lcol = (tid & 7) * 4;     // 0,4,..28

  v8f acc[2][4];
#pragma unroll
  for (int t = 0; t < 2; ++t)
#pragma unroll
    for (int u = 0; u < 4; ++u) acc[t][u] = (v8f)0.0f;

  // Stage one 128x32 slab of x and W into LDS buffer `buf`.
  auto stage = [&](int buf, int kt) {
#pragma unroll
    for (int p = 0; p < 4; ++p) {
      const int row = lrow + p * 32;
      const float* gx = x  + (size_t)(bm + row) * WIDTH_ + kt + lcol;
      const float* gw = Wl + (size_t)(bn + row) * WIDTH_ + kt + lcol;
#if USE_ASYNC
      cp_async16(gx, &sX[buf][row][lcol]);
      cp_async16(gw, &sW[buf][row][lcol]);
#else
      *reinterpret_cast<float4*>(&sX[buf][row][lcol]) =
          *reinterpret_cast<const float4*>(gx);
      *reinterpret_cast<float4*>(&sW[buf][row][lcol]) =
          *reinterpret_cast<const float4*>(gw);
#endif
    }
  };

  stage(0, 0);   // prologue: 8 async ops in flight per wave

  for (int kt = 0; kt < WIDTH_; kt += 32) {
    const int cur = (kt >> 5) & 1;
    if (kt + 32 < WIDTH_) stage(cur ^ 1, kt + 32);   // prefetch next slab
#if USE_ASYNC
    if (kt + 32 < WIDTH_) {
      __builtin_amdgcn_s_wait_asynccnt(8);   // drain current slab, keep next in flight
    } else {
      __builtin_amdgcn_s_wait_asynccnt(0);
    }
#endif
    __syncthreads();   // all waves' slab data visible

#pragma unroll
    for (int kk = 0; kk < 32; kk += 4) {
      const int k0 = kk + 2 * hh;
      v2f af[2], bf[4];
#pragma unroll
      for (int t = 0; t < 2; ++t) af[t] = ld2(&sX[cur][mw + t * 16 + l16][k0]);
#pragma unroll
      for (int u = 0; u < 4; ++u) bf[u] = ld2(&sW[cur][nw + u * 16 + l16][k0]);
#pragma unroll
      for (int t = 0; t < 2; ++t)
#pragma unroll
        for (int u = 0; u < 4; ++u)
          acc[t][u] = wmma_f32(af[t], bf[u], acc[t][u]);
    }
    __syncthreads();   // done reading `cur` before anyone overwrites it
  }

  // Rank-16 low-rank correction: acc += y(16-wide K) * Al^T, straight from L2.
#pragma unroll
  for (int kc = 0; kc < RANK_; kc += 4) {
    const int k0 = kc + 2 * hh;
    v2f af[2], bf[4];
#pragma unroll
    for (int t = 0; t < 2; ++t)
      af[t] = ld2(y + (size_t)(bm + mw + t * 16 + l16) * RANK_ + k0);
#pragma unroll
    for (int u = 0; u < 4; ++u)
      bf[u] = ld2(Al + (size_t)(bn + nw + u * 16 + l16) * RANK_ + k0);
#pragma unroll
    for (int t = 0; t < 2; ++t)
#pragma unroll
      for (int u = 0; u < 4; ++u)
        acc[t][u] = wmma_f32(af[t], bf[u], acc[t][u]);
  }

  // Epilogue: add bias, store. C layout: M = v + 8*half, N = l16.
#pragma unroll
  for (int t = 0; t < 2; ++t) {
#pragma unroll
    for (int u = 0; u < 4; ++u) {
      const int col = bn + nw + u * 16 + l16;
      const float bv = bias[col];
#pragma unroll
      for (int v = 0; v < 8; ++v) {
        const int row = bm + mw + t * 16 + 8 * hh + v;
        out[(size_t)row * NH_ + col] = acc[t][u][v] + bv;
      }
    }
  }
}

// ---------------------------------------------------------------------------
extern "C" void kernel_launch(void* const* d_in, const int* in_sizes, int n_in,
                              void* d_out, int out_size, void* d_ws, size_t ws_size,
                              hipStream_t stream) {
  const float* mu = (const float*)d_in[0];
  const float* nu = (const float*)d_in[1];
  const float* tt = (const float*)d_in[2];
  const float* Ws = (const float*)d_in[3];   // L x NH x WIDTH
  const float* As = (const float*)d_in[4];   // L x NH x RANK
  const float* Bs = (const float*)d_in[5];   // L x RANK x WIDTH
  const float* bs = (const float*)d_in[6];   // L x NH
  const float* aW = (const float*)d_in[7];   // L x 4
  const float* ab = (const float*)d_in[8];   // L
  float* out = (float*)d_out;

  // ws layout: [y: B*RANK floats][xbuf: B*WIDTH floats]  (~32.5 MB total)
  float* y    = (float*)d_ws;
  float* xbuf = (float*)((char*)d_ws + (size_t)B_TOT * RANK_ * sizeof(float));

  const float* cur = mu;
  for (int i = 0; i < L_; ++i) {
    float* nxt = (i & 1) ? out : xbuf;   // mu -> ws -> out -> ws -> out
    lowrank_kernel<<<dim3(B_TOT / 128), 256, 0, stream>>>(
        cur, Bs + (size_t)i * RANK_ * WIDTH_, nu, tt, aW + i * 4, ab + i, y);
    layer_gemm_kernel<<<dim3(B_TOT / 128, NH_ / 128), 256, 0, stream>>>(
        cur, Ws + (size_t)i * NH_ * WIDTH_, y, As + (size_t)i * NH_ * RANK_,
        bs + (size_t)i * NH_, nxt);
    cur = nxt;
  }
}